// E63mRNNCell_4887672783113
// MI455X (gfx1250) — compile-verified
//
#include <hip/hip_runtime.h>
#include <hip/hip_bf16.h>
#include <math.h>

// Problem dims (fixed by the reference)
#define T_ 512
#define B_ 8
#define D_ 256
#define N_ 256
#define MROWS (T_ * B_)      // 4096 rows of x
#define PITCH 258            // LDS row pitch (floats): pitch%64==2 -> conflict-free, 8B aligned rows

typedef float v2f __attribute__((ext_vector_type(2)));
typedef float v4f __attribute__((ext_vector_type(4)));
typedef float v8f __attribute__((ext_vector_type(8)));

// ---------------------------------------------------------------------------
// Kernel A: the four input projections via fp32 WMMA (V_WMMA_F32_16X16X4_F32)
//   out[which][m][e] = sum_d x[m][d] * W[e][d]  (+ bias[e] for which==2,3)
//   which: 0 -> k_all (W_k), 1 -> q_all (W_q), 2 -> Wx_all + b, 3 -> ax_all + b_alpha
// One wave computes one 16x16 output tile, accumulating K=256 in steps of 4.
// ---------------------------------------------------------------------------
__global__ __launch_bounds__(256) void proj_gemm_wmma(
    const float* __restrict__ x,
    const float* __restrict__ Wk, const float* __restrict__ Wq,
    const float* __restrict__ Wx, const float* __restrict__ Wa,
    const float* __restrict__ bv, const float* __restrict__ ba,
    float* __restrict__ out)
{
    const int which = blockIdx.y;                       // 0..3 (uniform)
    const float* __restrict__ W =
        (which == 0) ? Wk : (which == 1) ? Wq : (which == 2) ? Wx : Wa;
    const float* __restrict__ bias =
        (which == 2) ? bv : (which == 3) ? ba : nullptr;
    float* __restrict__ o = out + (size_t)which * MROWS * N_;

    const int wave = threadIdx.x >> 5;
    const int lane = threadIdx.x & 31;
    const int tile = blockIdx.x * 8 + wave;             // 4096 tiles total
    const int m0 = (tile >> 4) * 16;                    // 256 M-tiles
    const int n0 = (tile & 15) * 16;                    // 16  N-tiles
    const int lm = lane & 15;                           // M (for A) / N (for B) within tile
    const int lk = (lane >> 4) * 2;                     // K pair base per ISA 16x4 layout

    // A element (vgpr j, lane l): M = l%16, K = 2*(l/16) + j  -> contiguous f32 pair
    const float* __restrict__ ap = x + (size_t)(m0 + lm) * D_ + lk;
    // B element (vgpr j, lane l): N = l%16, K = 2*(l/16) + j ; B[k][n] = W[n][k]
    const float* __restrict__ bp = W + (size_t)(n0 + lm) * D_ + lk;

    v8f c = {};
    #pragma unroll 8
    for (int k0 = 0; k0 < D_; k0 += 4) {
        v2f a = *(const v2f*)(ap + k0);
        v2f b = *(const v2f*)(bp + k0);
        // 8 args: (neg_a, A, neg_b, B, c_mod, C, reuse_a, reuse_b)
        c = __builtin_amdgcn_wmma_f32_16x16x4_f32(
                false, a, false, b, (short)0, c, false, false);
    }

    const float badd = bias ? bias[n0 + lm] : 0.0f;
    // C/D layout: vgpr r, lane l: M = r + 8*(l/16), N = l%16
    const int rowb = m0 + (lane >> 4) * 8;
    const int col  = n0 + lm;
    #pragma unroll
    for (int r = 0; r < 8; ++r)
        o[(size_t)(rowb + r) * N_ + col] = c[r] + badd;
}

// ---------------------------------------------------------------------------
// Kernel B: sequential scan, one workgroup per batch, S resident in LDS.
// 256 threads; thread n owns state row S[n][0..255].
// ---------------------------------------------------------------------------
__global__ __launch_bounds__(256) void scan_kernel(
    const float* __restrict__ S0, const float* __restrict__ y0,
    const float* __restrict__ Wr, const float* __restrict__ R,
    const float* __restrict__ kall, const float* __restrict__ qall,
    const float* __restrict__ wxall, const float* __restrict__ axall,
    float* __restrict__ yout, float* __restrict__ Sout)
{
    __shared__ float S[N_ * PITCH];      // ~258 KB of the 320 KB WGP LDS
    __shared__ float ks[N_];
    __shared__ float qs[N_];
    __shared__ float retr[N_];
    __shared__ float ys[N_];

    const int b   = blockIdx.x;
    const int tid = threadIdx.x;

    // ---- init: S[0] = S0 (also emit output slab t=0), y_s = y0 ----
    const float* __restrict__ s0p = S0 + (size_t)b * N_ * D_;
    float* __restrict__ so0 = Sout + (size_t)b * N_ * D_;
    for (int n = 0; n < N_; ++n) {
        float v = s0p[(size_t)n * D_ + tid];     // coalesced: lanes span d
        S[n * PITCH + tid] = v;
        __builtin_nontemporal_store(v, so0 + (size_t)n * D_ + tid);
    }
    ys[tid] = y0[b * N_ + tid];
    __syncthreads();

    float* __restrict__ srow = &S[tid * PITCH];
    const float* __restrict__ wrrow = Wr + (size_t)tid * N_;
    const float* __restrict__ rrow  = R  + (size_t)tid * N_;

    for (int t = 0; t < T_; ++t) {
        const size_t base = ((size_t)t * B_ + b) * (size_t)N_;
        ks[tid] = kall[base + tid];
        qs[tid] = qall[base + tid];
        const float wx = wxall[base + tid];      // already includes +b
        const float ax = axall[base + tid];      // already includes +b_alpha
        __syncthreads();                         // ks/qs ready; prev store pass done

        // retrieved[n] = tanh( dot(S[n,:], k) )   (bank-conflict-free: pitch%64==2)
        float acc = 0.0f;
        #pragma unroll 8
        for (int d = 0; d < D_; d += 2) {
            acc = fmaf(srow[d],     ks[d],     acc);
            acc = fmaf(srow[d + 1], ks[d + 1], acc);
        }
        retr[tid] = tanhf(acc);
        __syncthreads();                         // retr published

        // v[n] = tanh( retr . Wr[n,:] + Wx + b ) ; rows of Wr stay hot in L2
        float vacc = wx;
        const float4* __restrict__ wr4 = (const float4*)wrrow;
        #pragma unroll 4
        for (int m4 = 0; m4 < N_ / 4; ++m4) {
            float4 w = wr4[m4];
            int m = m4 * 4;
            vacc = fmaf(w.x, retr[m],     vacc);
            vacc = fmaf(w.y, retr[m + 1], vacc);
            vacc = fmaf(w.z, retr[m + 2], vacc);
            vacc = fmaf(w.w, retr[m + 3], vacc);
        }
        const float v     = tanhf(vacc);
        const float alpha = 1.0f / (1.0f + __expf(-ax));
        const float omav  = (1.0f - alpha) * v;

        // S[n,:] = alpha*S[n,:] + (1-alpha)*v*k ; fused y-dot with q
        float yacc = 0.0f;
        #pragma unroll 4
        for (int d = 0; d < D_; d += 2) {
            float s0n = fmaf(alpha, srow[d],     omav * ks[d]);
            float s1n = fmaf(alpha, srow[d + 1], omav * ks[d + 1]);
            srow[d]     = s0n;
            srow[d + 1] = s1n;
            yacc = fmaf(s0n, qs[d],     yacc);
            yacc = fmaf(s1n, qs[d + 1], yacc);
        }

        // y recurrence: + y_prev . R[n,:]
        float racc = 0.0f;
        const float4* __restrict__ r4 = (const float4*)rrow;
        #pragma unroll 4
        for (int m4 = 0; m4 < N_ / 4; ++m4) {
            float4 w = r4[m4];
            int m = m4 * 4;
            racc = fmaf(w.x, ys[m],     racc);
            racc = fmaf(w.y, ys[m + 1], racc);
            racc = fmaf(w.z, ys[m + 2], racc);
            racc = fmaf(w.w, ys[m + 3], racc);
        }
        const float ynew = tanhf(yacc + racc);
        __syncthreads();                         // all S rows updated, all ys readers done

        ys[tid] = ynew;
        yout[base + tid] = ynew;

        // Coalesced non-temporal streaming store of S_new (the 1 GB output stream)
        float* __restrict__ sop = Sout + ((size_t)(t + 1) * B_ + b) * (size_t)(N_ * D_);
        #pragma unroll 2
        for (int j = 0; j < 64; ++j) {
            int flat = j * 1024 + tid * 4;       // lanes contiguous -> coalesced b128
            int n = flat >> 8;
            int d = flat & 255;
            float2 a0 = *(const float2*)&S[n * PITCH + d];
            float2 a1 = *(const float2*)&S[n * PITCH + d + 2];
            v4f o4 = {a0.x, a0.y, a1.x, a1.y};
            __builtin_nontemporal_store(o4, (v4f*)(sop + flat));
        }
        // next iteration's first __syncthreads closes the store-pass race
    }
}

// ---------------------------------------------------------------------------
extern "C" void kernel_launch(void* const* d_in, const int* in_sizes, int n_in,
                              void* d_out, int out_size, void* d_ws, size_t ws_size,
                              hipStream_t stream) {
    (void)in_sizes; (void)n_in; (void)out_size; (void)ws_size;
    const float* x   = (const float*)d_in[0];   // (T,B,D)
    const float* S0  = (const float*)d_in[1];   // (B,N,D)
    const float* y0  = (const float*)d_in[2];   // (B,N)
    const float* Wk  = (const float*)d_in[3];   // (D,D)
    const float* Wq  = (const float*)d_in[4];   // (D,D)
    const float* Wx  = (const float*)d_in[5];   // (N,D)
    const float* Wr  = (const float*)d_in[6];   // (N,N)
    const float* bb  = (const float*)d_in[7];   // (N,)
    const float* Wa  = (const float*)d_in[8];   // (N,D)
    const float* ba  = (const float*)d_in[9];   // (N,)
    const float* R   = (const float*)d_in[10];  // (N,N)

    float* yout = (float*)d_out;                       // (T,B,N)
    float* Sout = yout + (size_t)T_ * B_ * N_;         // (T+1,B,N,D)

    float* ws   = (float*)d_ws;                        // 4 x 4 MB projections
    float* kall = ws + (size_t)0 * MROWS * N_;
    float* qall = ws + (size_t)1 * MROWS * N_;
    float* wxa  = ws + (size_t)2 * MROWS * N_;
    float* axa  = ws + (size_t)3 * MROWS * N_;

    // Phase 1: WMMA projections (4096 tiles/weight, 8 waves per block)
    dim3 gA(MROWS / 16 * (N_ / 16) / 8, 4, 1);         // (512, 4)
    proj_gemm_wmma<<<gA, 256, 0, stream>>>(x, Wk, Wq, Wx, Wa, bb, ba, ws);

    // Phase 2: sequential scan, one WGP per batch with LDS-resident state
    scan_kernel<<<dim3(B_), 256, 0, stream>>>(S0, y0, Wr, R,
                                              kall, qall, wxa, axa,
                                              yout, Sout);
}